// Attention_Layer_18090402250757
// MI455X (gfx1250) — compile-verified
//
#include <hip/hip_runtime.h>

// CDNA5 / gfx1250, wave32. One wave per (b,l) token: B*L = 8192 tokens.
// - All global traffic b128-coalesced (512B/wave/instruction).
// - Fg reductions: wave32 ds butterflies; pre-exp tile stashed in VGPRs.
// - K-normalizer: chained V_WMMA_F32_16X16X4_F32 (A = ones), exact f32,
//   layout-invariant placement, D replicated so c[0] is the result.
// - node_outputs staged via GLOBAL_LOAD_ASYNC_TO_LDS_B128 (ASYNCcnt);
//   copy latency hidden behind ~1K instructions of softmax math.

typedef __attribute__((ext_vector_type(2))) float v2f;
typedef __attribute__((ext_vector_type(4))) float v4f;
typedef __attribute__((ext_vector_type(8))) float v8f;

typedef int v4i __attribute__((vector_size(16)));          // matches builtin
typedef __attribute__((address_space(1))) v4i* as1_v4i;    // global (AS1)
typedef __attribute__((address_space(3))) v4i* as3_v4i;    // LDS (AS3)

#define AEPS 1e-6f

#if __has_builtin(__builtin_amdgcn_global_load_async_to_lds_b128)
#define USE_ASYNC_LDS 1
#else
#define USE_ASYNC_LDS 0
#endif

__global__ __launch_bounds__(256, 1) void gat_attn_kernel(
    const float* __restrict__ beta,       // [T,128]  (Fg=32,H=4)
    const float* __restrict__ self_attn,  // [T,128]
    const float* __restrict__ attn_coef,  // [T,16,128]
    const float* __restrict__ node_out,   // [T,16,128] (Fo=32,H=4)
    const float* __restrict__ graph_w,    // [T,16,32]
    float* __restrict__ out,              // [T,128]
    float* __restrict__ fin_out,          // [T,16,4]
    int ntok)
{
  __shared__ v4f fin_s[8][16];            // per-wave raw fin[k][h]
#if USE_ASYNC_LDS
  __shared__ v4f no_s[8][16 * 32];        // per-wave staged node_outputs (8KB)
#endif

  const int lane  = threadIdx.x & 31;
  const int wv    = threadIdx.x >> 5;
  const int token = blockIdx.x * 8 + wv;
  if (token >= ntok) return;   // uniform per wave; WMMA below has full EXEC

  const v4f*   b4p = (const v4f*)(beta      + (size_t)token * 128);
  const v4f*   s4p = (const v4f*)(self_attn + (size_t)token * 128);
  const v4f*   acp = (const v4f*)(attn_coef + (size_t)token * 2048);
  const v4f*   nop = (const v4f*)(node_out  + (size_t)token * 2048);
  const float* gwp = graph_w + (size_t)token * 512;

#if USE_ASYNC_LDS
  // Kick off the phase-4 stream now: 16 x (512B/wave) async b128 transfers,
  // tracked by ASYNCcnt, consumed only after the softmax/normalizer work.
  #pragma unroll
  for (int k = 0; k < 16; ++k) {
    __builtin_amdgcn_global_load_async_to_lds_b128(
        (as1_v4i)(nop + k * 32 + lane),
        (as3_v4i)(&no_s[wv][k * 32 + lane]),
        0, 0);
  }
#else
  __builtin_prefetch(nop + lane * 16, 0, 1);   // global_prefetch_b8 fallback
#endif
  __builtin_prefetch(acp + lane, 0, 1);

  // lane t owns j = 4t..4t+3 of every 128-wide row; component c == h
  v4f b4 = b4p[lane] + AEPS;    // (beta + eps); layout fg*4+h matches
  v4f s4 = s4p[lane];

  const float ninf = -__builtin_inff();
  v4f mx = {ninf, ninf, ninf, ninf};
  v4f v[16];                    // pre-exp stash: 64 VGPRs, no HBM re-read

  #pragma unroll
  for (int k = 0; k < 16; ++k) {
    v4f a = acp[k * 32 + lane];               // global_load_b128
    if (k == 0) a += s4;                      // self-attn into neighbor slot 0
    a *= b4;
    v[k] = a;
    mx.x = fmaxf(mx.x, a.x); mx.y = fmaxf(mx.y, a.y);
    mx.z = fmaxf(mx.z, a.z); mx.w = fmaxf(mx.w, a.w);
  }
  // per-h max over (K, Fg): Fg lives across the 32 lanes -> butterfly
  #pragma unroll
  for (int off = 16; off > 0; off >>= 1) {
    mx.x = fmaxf(mx.x, __shfl_xor(mx.x, off, 32));
    mx.y = fmaxf(mx.y, __shfl_xor(mx.y, off, 32));
    mx.z = fmaxf(mx.z, __shfl_xor(mx.z, off, 32));
    mx.w = fmaxf(mx.w, __shfl_xor(mx.w, off, 32));
  }

  // fin[k,h] = sum_fg gw[k,fg] * exp(ac'[k,fg,h] - max_h)
  #pragma unroll
  for (int k = 0; k < 16; ++k) {
    float g = gwp[k * 32 + lane];             // coalesced b32 (fg = lane)
    v4f e;
    e.x = __expf(v[k].x - mx.x) * g;
    e.y = __expf(v[k].y - mx.y) * g;
    e.z = __expf(v[k].z - mx.z) * g;
    e.w = __expf(v[k].w - mx.w) * g;
    #pragma unroll
    for (int off = 16; off > 0; off >>= 1) {
      e.x += __shfl_xor(e.x, off, 32);
      e.y += __shfl_xor(e.y, off, 32);
      e.z += __shfl_xor(e.z, off, 32);
      e.w += __shfl_xor(e.w, off, 32);
    }
    if (lane == 0) fin_s[wv][k] = e;          // same-wave LDS, DS in-order
  }

  // ---- denom[h] = sum_k |fin[k,h]| via chained V_WMMA_F32_16X16X4_F32 ----
  // A = ones(16x4). One h per chain; each (lane-half x VGPR) group carries one
  // distinct k per step -> column sums are B-layout-invariant, D replicated.
  const float* fs = (const float*)(&fin_s[wv][0]);
  const int kadd = (lane >= 16) ? 2 : 0;
  v2f ones = {1.0f, 1.0f};
  float den[4];
  #pragma unroll
  for (int h = 0; h < 4; ++h) {
    v8f c = {0.f, 0.f, 0.f, 0.f, 0.f, 0.f, 0.f, 0.f};
    #pragma unroll
    for (int s = 0; s < 4; ++s) {
      const int k0 = 4 * s + kadd;
      v2f bb;
      bb.x = fabsf(fs[k0 * 4 + h]);
      bb.y = fabsf(fs[(k0 + 1) * 4 + h]);
      c = __builtin_amdgcn_wmma_f32_16x16x4_f32(
          /*neg_a=*/false, ones, /*neg_b=*/false, bb,
          /*c_mod=*/(short)0, c, /*reuse_a=*/false, /*reuse_b=*/false);
    }
    den[h] = c[0];
  }
  v4f inv;
  inv.x = 1.0f / (den[0] + AEPS);
  inv.y = 1.0f / (den[1] + AEPS);
  inv.z = 1.0f / (den[2] + AEPS);
  inv.w = 1.0f / (den[3] + AEPS);

  // normalized fin output: [T,16,4], lanes 0..15 store 16B each (256B/wave)
  if (lane < 16) {
    v4f f = fin_s[wv][lane] * inv;
    ((v4f*)(fin_out + (size_t)token * 64))[lane] = f;
  }

#if USE_ASYNC_LDS
#if __has_builtin(__builtin_amdgcn_s_wait_asynccnt)
  __builtin_amdgcn_s_wait_asynccnt(0);
#else
  asm volatile("s_wait_asynccnt 0" ::: "memory");
#endif
#endif

  // out[j] = sum_k no[k,j] * fin_norm[k, j%4]; component c == h again
  v4f acc = {0.f, 0.f, 0.f, 0.f};
  #pragma unroll
  for (int k = 0; k < 16; ++k) {
#if USE_ASYNC_LDS
    v4f n4 = no_s[wv][k * 32 + lane];         // ds_load_b128, conflict-free
#else
    v4f n4 = nop[k * 32 + lane];              // global_load_b128
#endif
    v4f f  = fin_s[wv][k] * inv;              // ds_load_b128 broadcast
    acc.x = fmaf(n4.x, f.x, acc.x);
    acc.y = fmaf(n4.y, f.y, acc.y);
    acc.z = fmaf(n4.z, f.z, acc.z);
    acc.w = fmaf(n4.w, f.w, acc.w);
  }
  ((v4f*)(out + (size_t)token * 128))[lane] = acc;  // global_store_b128
}

extern "C" void kernel_launch(void* const* d_in, const int* in_sizes, int n_in,
                              void* d_out, int out_size, void* d_ws, size_t ws_size,
                              hipStream_t stream) {
  (void)n_in; (void)out_size; (void)d_ws; (void)ws_size;
  const float* beta = (const float*)d_in[0];
  const float* sa   = (const float*)d_in[1];
  const float* ac   = (const float*)d_in[2];
  const float* no   = (const float*)d_in[3];
  const float* gw   = (const float*)d_in[4];

  const int ntok = in_sizes[0] / 128;                 // B*L = 8192
  float* outp = (float*)d_out;                        // [T,128] then [T,16,4]
  float* finp = outp + (size_t)ntok * 128;

  const int blocks = (ntok + 7) / 8;                  // 8 waves (tokens)/block
  gat_attn_kernel<<<blocks, 256, 0, stream>>>(beta, sa, ac, no, gw, outp, finp, ntok);
}